// LocalAttention_72533407695115
// MI455X (gfx1250) — compile-verified
//
#include <hip/hip_runtime.h>

typedef float v2f __attribute__((ext_vector_type(2)));
typedef float v8f __attribute__((ext_vector_type(8)));

#define N_PTS    2048
#define C_CH     256
#define H_HEADS  8
#define DK_DIM   32
#define RADIUS_F 0.1f
#define MASK_NEG -1.0e30f
#define HNDK     (N_PTS * DK_DIM)          // 65536 floats per head plane
#define QKV_ONE  (H_HEADS * N_PTS * DK_DIM) // 524288 floats per q/k/v

// ---- row reductions: rows of a 16x16 WMMA C/D tile live in 16-lane halves ----
__device__ __forceinline__ float rowmax16(float x) {
#pragma unroll
  for (int m = 1; m < 16; m <<= 1) x = fmaxf(x, __shfl_xor(x, m, 16));
  return x;
}
__device__ __forceinline__ float rowsum16(float x) {
#pragma unroll
  for (int m = 1; m < 16; m <<= 1) x += __shfl_xor(x, m, 16);
  return x;
}

// ======================= Kernel 1: QKV pointwise GEMM =======================
// out[768][2048] = W[768][256] @ X[256][2048] + b ; scattered to q/k/v planes
// One wave per 16x16 output tile, fp32 WMMA, K stepped by 4.
__global__ __launch_bounds__(32) void qkv_gemm_k(const float* __restrict__ W,
                                                 const float* __restrict__ X,
                                                 const float* __restrict__ bias,
                                                 float* __restrict__ out) {
  const int lane = threadIdx.x;
  const int half = lane >> 4;
  const int lcol = lane & 15;
  const int o0 = blockIdx.x * 16;   // 0..767 (48 tiles)
  const int n0 = blockIdx.y * 16;   // 0..2047 (128 tiles)

  v8f acc = {};
#pragma unroll 4
  for (int k = 0; k < C_CH; k += 4) {
    // A frag: lane holds row M=lcol, K = k + {0,1} (low half) / {2,3} (high half)
    v2f a = *(const v2f*)(W + (o0 + lcol) * C_CH + k + 2 * half);
    // B frag: col N=lcol; VGPR0 row k+2*half, VGPR1 row k+2*half+1
    v2f b;
    b.x = X[(k + 2 * half) * N_PTS + n0 + lcol];
    b.y = X[(k + 2 * half + 1) * N_PTS + n0 + lcol];
    acc = __builtin_amdgcn_wmma_f32_16x16x4_f32(false, a, false, b, (short)0,
                                                acc, false, false);
  }
#pragma unroll
  for (int r = 0; r < 8; ++r) {
    const int o = o0 + r + 8 * half;          // D row M = r + 8*half
    const float val = acc[r] + bias[o];
    const int which = o >> 8;                  // 0=q 1=k 2=v
    const int rem = o & 255;
    const int h = rem >> 5, dk = rem & 31;     // head, channel
    out[which * QKV_ONE + h * HNDK + (n0 + lcol) * DK_DIM + dk] = val;
  }
}

// =================== Kernel 2: fused rel-pos MLP + attention ===================
// One wave per (16-query tile, 4-head group). Flash softmax over 128 key tiles.
__global__ __launch_bounds__(32) void attn_k(const float* __restrict__ xyz,
                                             const float* __restrict__ w1,
                                             const float* __restrict__ b1,
                                             const float* __restrict__ w2,
                                             const float* __restrict__ b2,
                                             const float* __restrict__ qF,
                                             const float* __restrict__ kF,
                                             const float* __restrict__ vF,
                                             float* __restrict__ out) {
  __shared__ float w1_l[128];   // [in=4][32]
  __shared__ float b1_l[32];
  __shared__ float w2h_l[128];  // [j=32][hh=4]  (channel-mean folded, this head group)
  __shared__ float b2h_l[4];
  __shared__ float plds[256];   // 16x16 P tile for D->A re-fragmentation

  const int lane = threadIdx.x;
  const int half = lane >> 4;
  const int lcol = lane & 15;
  const int i0 = blockIdx.x * 16;  // query tile
  const int h0 = blockIdx.y * 4;   // head group base

  // ---- stage weights (fold mean over 4 pe channels into w2h/b2h) ----
#pragma unroll
  for (int t = 0; t < 4; ++t) w1_l[lane + 32 * t] = w1[lane + 32 * t];
  b1_l[lane] = b1[lane];
#pragma unroll
  for (int hh = 0; hh < 4; ++hh) {
    const float* p = w2 + lane * DK_DIM + 4 * (h0 + hh);
    w2h_l[lane * 4 + hh] = 0.25f * (p[0] + p[1] + p[2] + p[3]);
  }
  if (lane < 4) {
    const float* p = b2 + 4 * (h0 + lane);
    b2h_l[lane] = 0.25f * (p[0] + p[1] + p[2] + p[3]);
  }
  __syncthreads();

  // ---- query positions (row M = r + 8*half) ----
  float qx[8], qy[8], qz[8];
#pragma unroll
  for (int r = 0; r < 8; ++r) {
    const int qi = i0 + r + 8 * half;
    qx[r] = xyz[qi];
    qy[r] = xyz[N_PTS + qi];
    qz[r] = xyz[2 * N_PTS + qi];
  }

  // ---- Q A-fragments, resident for whole pass (4 heads x 8 K-steps) ----
  v2f qa[4][8];
#pragma unroll
  for (int h = 0; h < 4; ++h)
#pragma unroll
    for (int s = 0; s < 8; ++s)
      qa[h][s] = *(const v2f*)(qF + (h0 + h) * HNDK + (i0 + lcol) * DK_DIM +
                               4 * s + 2 * half);

  v8f oacc[4][2];
  float mrun[4][8], lrun[4][8];
#pragma unroll
  for (int h = 0; h < 4; ++h) {
    oacc[h][0] = {};
    oacc[h][1] = {};
#pragma unroll
    for (int r = 0; r < 8; ++r) { mrun[h][r] = MASK_NEG; lrun[h][r] = 0.0f; }
  }

  for (int m0 = 0; m0 < N_PTS; m0 += 16) {
    // hint next key/value tiles into cache
    __builtin_prefetch(kF + h0 * HNDK + (m0 + 16) * DK_DIM, 0, 3);
    __builtin_prefetch(vF + h0 * HNDK + (m0 + 16) * DK_DIM, 0, 3);

    // ---- rel-pos MLP + mask for this 16x16 pair tile (shared by 4 heads) ----
    const float kx = xyz[m0 + lcol];
    const float ky = xyz[N_PTS + m0 + lcol];
    const float kz = xyz[2 * N_PTS + m0 + lcol];
    float pe[8][4], dist[8];
#pragma unroll
    for (int r = 0; r < 8; ++r) {
      const float dx = qx[r] - kx, dy = qy[r] - ky, dz = qz[r] - kz;
      const float d = sqrtf(dx * dx + dy * dy + dz * dz);
      dist[r] = d;
      float p0 = b2h_l[0], p1 = b2h_l[1], p2 = b2h_l[2], p3 = b2h_l[3];
#pragma unroll 4
      for (int j = 0; j < 32; ++j) {
        float hid = b1_l[j] + dx * w1_l[j] + dy * w1_l[32 + j] +
                    dz * w1_l[64 + j] + d * w1_l[96 + j];
        hid = fmaxf(hid, 0.0f);
        p0 += hid * w2h_l[j * 4 + 0];
        p1 += hid * w2h_l[j * 4 + 1];
        p2 += hid * w2h_l[j * 4 + 2];
        p3 += hid * w2h_l[j * 4 + 3];
      }
      pe[r][0] = p0; pe[r][1] = p1; pe[r][2] = p2; pe[r][3] = p3;
    }

#pragma unroll
    for (int h = 0; h < 4; ++h) {
      const float* kbase = kF + (h0 + h) * HNDK;
      const float* vbase = vF + (h0 + h) * HNDK;

      // ---- S = Q * K^T  (16x16, K-dim = 32 over 8 steps) ----
      v8f s = {};
#pragma unroll
      for (int st = 0; st < 8; ++st) {
        v2f b = *(const v2f*)(kbase + (m0 + lcol) * DK_DIM + 4 * st + 2 * half);
        s = __builtin_amdgcn_wmma_f32_16x16x4_f32(false, qa[h][st], false, b,
                                                  (short)0, s, false, false);
      }

      // ---- +pe, radius mask, online softmax (per-row stats in 16-lane halves) ----
      float p[8], scl[8];
#pragma unroll
      for (int r = 0; r < 8; ++r) {
        const float sv = (dist[r] < RADIUS_F) ? (s[r] + pe[r][h]) : MASK_NEG;
        const float tmax = rowmax16(sv);
        const float mnew = fmaxf(mrun[h][r], tmax);
        scl[r] = __expf(mrun[h][r] - mnew);
        const float pv = __expf(sv - mnew);
        lrun[h][r] = lrun[h][r] * scl[r] + rowsum16(pv);
        mrun[h][r] = mnew;
        p[r] = pv;
      }
#pragma unroll
      for (int r = 0; r < 8; ++r) {
        oacc[h][0][r] *= scl[r];
        oacc[h][1][r] *= scl[r];
      }

      // ---- re-fragment P (D-layout) -> A-layout via LDS ----
      __syncthreads();
#pragma unroll
      for (int r = 0; r < 8; ++r) plds[(r + 8 * half) * 16 + lcol] = p[r];
      __syncthreads();

      // ---- O += P * V  (K-dim = 16 keys over 4 steps, two 16-col tiles) ----
#pragma unroll
      for (int st = 0; st < 4; ++st) {
        v2f a;
        a.x = plds[lcol * 16 + 4 * st + 2 * half];
        a.y = plds[lcol * 16 + 4 * st + 2 * half + 1];
#pragma unroll
        for (int t = 0; t < 2; ++t) {
          v2f b;
          b.x = vbase[(m0 + 4 * st + 2 * half) * DK_DIM + 16 * t + lcol];
          b.y = vbase[(m0 + 4 * st + 2 * half + 1) * DK_DIM + 16 * t + lcol];
          oacc[h][t] = __builtin_amdgcn_wmma_f32_16x16x4_f32(
              false, a, false, b, (short)0, oacc[h][t], false, false);
        }
      }
    }
  }

  // ---- finalize: divide by row sums, write scrambled [n][h][dk] -> [C][N] ----
#pragma unroll
  for (int h = 0; h < 4; ++h)
#pragma unroll
    for (int t = 0; t < 2; ++t)
#pragma unroll
      for (int r = 0; r < 8; ++r) {
        const float val = oacc[h][t][r] / lrun[h][r];
        out[(i0 + r + 8 * half) * C_CH + (h0 + h) * DK_DIM + 16 * t + lcol] = val;
      }
}

extern "C" void kernel_launch(void* const* d_in, const int* in_sizes, int n_in,
                              void* d_out, int out_size, void* d_ws,
                              size_t ws_size, hipStream_t stream) {
  const float* xyz   = (const float*)d_in[0];
  const float* feats = (const float*)d_in[1];
  const float* qkv_w = (const float*)d_in[2];
  const float* qkv_b = (const float*)d_in[3];
  const float* w1    = (const float*)d_in[4];
  const float* b1    = (const float*)d_in[5];
  const float* w2    = (const float*)d_in[6];
  const float* b2    = (const float*)d_in[7];
  float* out = (float*)d_out;
  float* ws  = (float*)d_ws;  // needs 3*8*2048*32*4 = 6 MB

  float* qF = ws;
  float* kF = ws + QKV_ONE;
  float* vF = ws + 2 * QKV_ONE;

  // 768/16 o-tiles x 2048/16 n-tiles, one wave each
  qkv_gemm_k<<<dim3(48, 128), 32, 0, stream>>>(qkv_w, feats, qkv_b, ws);
  // 128 query tiles x 2 head groups (4 heads each), one wave each
  attn_k<<<dim3(128, 2), 32, 0, stream>>>(xyz, w1, b1, w2, b2, qF, kF, vF, out);
}